// SE_GNN_60395830116399
// MI455X (gfx1250) — compile-verified
//
#include <hip/hip_runtime.h>
#include <math.h>

#define N_NODES 50000
#define N_EDGES 1000000
#define DIM     128
#define BATCH   4096
#define NQ      (2 * BATCH)   // 8192 query rows

typedef __attribute__((ext_vector_type(2))) float v2f;
typedef __attribute__((ext_vector_type(8))) float v8f;

// ---------- float atomic min/max via sign-split integer atomics ----------
__device__ __forceinline__ void atomicMaxF32(float* addr, float val) {
    if (val >= 0.0f) atomicMax((int*)addr, __float_as_int(val));
    else             atomicMin((unsigned int*)addr, __float_as_uint(val));
}
__device__ __forceinline__ void atomicMinF32(float* addr, float val) {
    if (val >= 0.0f) atomicMin((int*)addr, __float_as_int(val));
    else             atomicMax((unsigned int*)addr, __float_as_uint(val));
}

// ---------- generic fill ----------
__global__ void fill_f32(float* __restrict__ p, float v, int n) {
    int i = blockIdx.x * blockDim.x + threadIdx.x;
    if (i < n) p[i] = v;
}

// ---------- column min/max of ent_emb ----------
__global__ void colminmax_kernel(const float* __restrict__ ent,
                                 float* __restrict__ xmin, float* __restrict__ xmax) {
    int c = threadIdx.x;                       // 0..127
    int rows_per_blk = (N_NODES + gridDim.x - 1) / gridDim.x;
    int r0 = blockIdx.x * rows_per_blk;
    int r1 = min(N_NODES, r0 + rows_per_blk);
    float mn = INFINITY, mx = -INFINITY;
    for (int r = r0; r < r1; ++r) {
        float v = ent[(long)r * DIM + c];
        mn = fminf(mn, v); mx = fmaxf(mx, v);
    }
    if (r1 > r0) { atomicMinF32(&xmin[c], mn); atomicMaxF32(&xmax[c], mx); }
}

// ---------- pass 1: per-edge logits + segment max (one wave32 per edge) ----------
__global__ __launch_bounds__(256)
void edge_logits_max(const float* __restrict__ ent, const float* __restrict__ rel,
                     const int* __restrict__ src, const int* __restrict__ dst,
                     const int* __restrict__ rid,
                     float* __restrict__ lg_e, float* __restrict__ lg_n, float* __restrict__ lg_c,
                     float* __restrict__ m_e,  float* __restrict__ m_n,  float* __restrict__ m_c) {
    int lane = threadIdx.x & 31;
    int e = blockIdx.x * 8 + (threadIdx.x >> 5);
    if (e >= N_EDGES) return;
    int s = src[e], d = dst[e], r = rid[e];
    float4 a = ((const float4*)(ent + (long)s * DIM))[lane];   // h_src (4 of 128)
    float4 b = ((const float4*)(ent + (long)d * DIM))[lane];   // h_dst
    float4 c = ((const float4*)(rel + (long)r * DIM))[lane];   // r
    float pe = c.x*b.x + c.y*b.y + c.z*b.z + c.w*b.w;                         // r . h_dst
    float pn = a.x*b.x + a.y*b.y + a.z*b.z + a.w*b.w;                         // h_src . h_dst
    float pc = a.x*c.x*b.x + a.y*c.y*b.y + a.z*c.z*b.z + a.w*c.w*b.w;         // (h_src*r) . h_dst
    #pragma unroll
    for (int off = 16; off; off >>= 1) {
        pe += __shfl_xor(pe, off, 32);
        pn += __shfl_xor(pn, off, 32);
        pc += __shfl_xor(pc, off, 32);
    }
    if (lane == 0) {
        lg_e[e] = pe; lg_n[e] = pn; lg_c[e] = pc;
        atomicMaxF32(&m_e[d], pe);
        atomicMaxF32(&m_n[d], pn);
        atomicMaxF32(&m_c[d], pc);
    }
}

// ---------- pass 2: exp, denominators, weighted scatter (atomics land in L2) ----------
__global__ __launch_bounds__(256)
void edge_accum(const float* __restrict__ ent, const float* __restrict__ rel,
                const int* __restrict__ src, const int* __restrict__ dst,
                const int* __restrict__ rid,
                const float* __restrict__ lg_e, const float* __restrict__ lg_n, const float* __restrict__ lg_c,
                const float* __restrict__ m_e,  const float* __restrict__ m_n,  const float* __restrict__ m_c,
                float* __restrict__ s_e, float* __restrict__ s_n, float* __restrict__ s_c,
                float* __restrict__ ne,  float* __restrict__ nn,  float* __restrict__ nc) {
    int lane = threadIdx.x & 31;
    int e = blockIdx.x * 8 + (threadIdx.x >> 5);
    if (e >= N_EDGES) return;
    int s = src[e], d = dst[e], r = rid[e];
    float we = expf(lg_e[e] - m_e[d]);   // uniform address per wave -> broadcast load
    float wn = expf(lg_n[e] - m_n[d]);
    float wc = expf(lg_c[e] - m_c[d]);
    if (lane == 0) {
        atomicAdd(&s_e[d], we); atomicAdd(&s_n[d], wn); atomicAdd(&s_c[d], wc);
    }
    float4 a = ((const float4*)(ent + (long)s * DIM))[lane];   // h_src
    float4 c = ((const float4*)(rel + (long)r * DIM))[lane];   // r
    float* pe = ne + (long)d * DIM + lane * 4;
    float* pn = nn + (long)d * DIM + lane * 4;
    float* pc = nc + (long)d * DIM + lane * 4;
    atomicAdd(pe + 0, c.x * we); atomicAdd(pe + 1, c.y * we);
    atomicAdd(pe + 2, c.z * we); atomicAdd(pe + 3, c.w * we);
    atomicAdd(pn + 0, a.x * wn); atomicAdd(pn + 1, a.y * wn);
    atomicAdd(pn + 2, a.z * wn); atomicAdd(pn + 3, a.w * wn);
    atomicAdd(pc + 0, a.x * c.x * wc); atomicAdd(pc + 1, a.y * c.y * wc);
    atomicAdd(pc + 2, a.z * c.z * wc); atomicAdd(pc + 3, a.w * c.w * wc);
}

// ---------- normalize neigh by softmax denominators ----------
__global__ void neigh_norm(float* __restrict__ ne, float* __restrict__ nn, float* __restrict__ nc,
                           const float* __restrict__ s_e, const float* __restrict__ s_n,
                           const float* __restrict__ s_c) {
    long i = (long)blockIdx.x * blockDim.x + threadIdx.x;
    if (i >= (long)N_NODES * DIM) return;
    int node = (int)(i >> 7);
    float se = s_e[node], sn = s_n[node], sc = s_c[node];
    ne[i] = se > 0.0f ? ne[i] / se : 0.0f;
    nn[i] = sn > 0.0f ? nn[i] / sn : 0.0f;
    nc[i] = sc > 0.0f ? nc[i] / sc : 0.0f;
}

// ---------- WMMA gather-GEMM (f32 16x16x4) + fused BN/tanh/minmax/residual ----------
// grid.x = NQ/16 = 512 blocks; 256 threads = 8 waves; wave w computes columns [16w,16w+16)
__global__ __launch_bounds__(256)
void wmma_final(const float* __restrict__ ne, const float* __restrict__ nn,
                const float* __restrict__ nc,
                const float* __restrict__ We, const float* __restrict__ Wn,
                const float* __restrict__ Wc,
                const float* __restrict__ ge, const float* __restrict__ be,
                const float* __restrict__ gn, const float* __restrict__ bn,
                const float* __restrict__ gc, const float* __restrict__ bc,
                const int* __restrict__ d1, const int* __restrict__ d2,
                const float* __restrict__ ent,
                const float* __restrict__ xmin, const float* __restrict__ xmax,
                float* __restrict__ out) {
    int lane = threadIdx.x & 31;
    int wave = threadIdx.x >> 5;              // 0..7 -> column tile
    int qbase = blockIdx.x * 16;              // 16 query rows per block
    int mrow  = lane & 15;                    // A-matrix row for this lane
    int khalf = (lane >> 4) << 1;             // K sub-offset 0 or 2
    int col   = wave * 16 + (lane & 15);      // global output column (B/C/D layout)

    int qiA = qbase + mrow;
    int idA = qiA < BATCH ? d1[qiA] : d2[qiA - BATCH];
    const float* Ae = ne + (long)idA * DIM;
    const float* An = nn + (long)idA * DIM;
    const float* Ac = nc + (long)idA * DIM;

    v8f acc_e = {0,0,0,0,0,0,0,0};
    v8f acc_n = {0,0,0,0,0,0,0,0};
    v8f acc_c = {0,0,0,0,0,0,0,0};

    for (int k = 0; k < DIM; k += 4) {
        int k0 = k + khalf;
        v2f a_e = *(const v2f*)(Ae + k0);
        v2f a_n = *(const v2f*)(An + k0);
        v2f a_c = *(const v2f*)(Ac + k0);
        v2f b_e = { We[k0 * DIM + col], We[(k0 + 1) * DIM + col] };
        v2f b_n = { Wn[k0 * DIM + col], Wn[(k0 + 1) * DIM + col] };
        v2f b_c = { Wc[k0 * DIM + col], Wc[(k0 + 1) * DIM + col] };
        acc_e = __builtin_amdgcn_wmma_f32_16x16x4_f32(false, a_e, false, b_e,
                                                      (short)0, acc_e, false, false);
        acc_n = __builtin_amdgcn_wmma_f32_16x16x4_f32(false, a_n, false, b_n,
                                                      (short)0, acc_n, false, false);
        acc_c = __builtin_amdgcn_wmma_f32_16x16x4_f32(false, a_c, false, b_c,
                                                      (short)0, acc_c, false, false);
    }

    // epilogue: BN(eval)+tanh per layer, min-max normalize, residual, gather-store
    const float isq = 1.0f / sqrtf(1.0f + 1e-5f);
    float sce = ge[col] * isq, scn = gn[col] * isq, scc = gc[col] * isq;
    float bbe = be[col],  bbn = bn[col],  bbc = bc[col];
    float mn  = xmin[col];
    float inv = 1.0f / (xmax[col] - mn);

    #pragma unroll
    for (int i = 0; i < 8; ++i) {
        int row = i + ((lane >> 4) << 3);     // C/D layout: VGPR i, lanes 16-31 hold M=i+8
        int qi  = qbase + row;
        int id  = qi < BATCH ? d1[qi] : d2[qi - BATCH];
        float ve = tanhf(acc_e[i] * sce + bbe);
        float vn = tanhf(acc_n[i] * scn + bbn);
        float vc = tanhf(acc_c[i] * scc + bbc);
        out[(long)qi * DIM + col] =
            ent[(long)id * DIM + col] + (ve - mn) * inv + (vn - mn) * inv + (vc - mn) * inv;
    }
}

extern "C" void kernel_launch(void* const* d_in, const int* in_sizes, int n_in,
                              void* d_out, int out_size, void* d_ws, size_t ws_size,
                              hipStream_t stream) {
    const int*   d1  = (const int*)d_in[0];
    const int*   d2  = (const int*)d_in[1];
    const int*   src = (const int*)d_in[2];
    const int*   dst = (const int*)d_in[3];
    const int*   rid = (const int*)d_in[4];
    const float* ent = (const float*)d_in[5];
    const float* rel = (const float*)d_in[6];
    const float* We  = (const float*)d_in[7];
    const float* Wn  = (const float*)d_in[8];
    const float* Wc  = (const float*)d_in[9];
    const float* ge  = (const float*)d_in[10];
    const float* be  = (const float*)d_in[11];
    const float* gn  = (const float*)d_in[12];
    const float* bn  = (const float*)d_in[13];
    const float* gc  = (const float*)d_in[14];
    const float* bc  = (const float*)d_in[15];
    float* out = (float*)d_out;

    // ---- workspace layout (floats) ----
    const long N = N_NODES, E = N_EDGES, ND = (long)N_NODES * DIM;
    float* ws   = (float*)d_ws;
    float* s_e  = ws;                 // 3N  : softmax denominators      (zero)
    float* s_n  = s_e + N;
    float* s_c  = s_n + N;
    float* ne   = s_c + N;            // 3ND : weighted neighbor sums    (zero)
    float* nn   = ne + ND;
    float* nc   = nn + ND;
    float* m_e  = nc + ND;            // 3N  : segment maxima            (-inf)
    float* m_n  = m_e + N;
    float* m_c  = m_n + N;
    float* lg_e = m_c + N;            // 3E  : edge logits               (written first)
    float* lg_n = lg_e + E;
    float* lg_c = lg_n + E;
    float* xmin = lg_c + E;           // D   : column min                (+inf)
    float* xmax = xmin + DIM;         // D   : column max                (-inf)

    long nzero = 3 * N + 3 * ND;
    fill_f32<<<(int)((nzero + 255) / 256), 256, 0, stream>>>(s_e, 0.0f, (int)nzero);
    fill_f32<<<(int)((3 * N + 255) / 256), 256, 0, stream>>>(m_e, -INFINITY, (int)(3 * N));
    fill_f32<<<1, 256, 0, stream>>>(xmin,  INFINITY, DIM);
    fill_f32<<<1, 256, 0, stream>>>(xmax, -INFINITY, DIM);

    colminmax_kernel<<<200, 128, 0, stream>>>(ent, xmin, xmax);

    // one wave32 per edge, 8 edges per 256-thread block
    int eblocks = (N_EDGES + 7) / 8;
    edge_logits_max<<<eblocks, 256, 0, stream>>>(ent, rel, src, dst, rid,
                                                 lg_e, lg_n, lg_c, m_e, m_n, m_c);
    edge_accum<<<eblocks, 256, 0, stream>>>(ent, rel, src, dst, rid,
                                            lg_e, lg_n, lg_c, m_e, m_n, m_c,
                                            s_e, s_n, s_c, ne, nn, nc);
    neigh_norm<<<(int)((ND + 255) / 256), 256, 0, stream>>>(ne, nn, nc, s_e, s_n, s_c);

    wmma_final<<<NQ / 16, 256, 0, stream>>>(ne, nn, nc, We, Wn, Wc,
                                            ge, be, gn, bn, gc, bc,
                                            d1, d2, ent, xmin, xmax, out);
}